// InverseDCT_1271310319945
// MI455X (gfx1250) — compile-verified
//
#include <hip/hip_runtime.h>
#include <hip/hip_bf16.h>

typedef float v2f __attribute__((ext_vector_type(2)));
typedef float v8f __attribute__((ext_vector_type(8)));

#define CH     64
#define HB     256
#define WB     256
#define NBATCH 16
#define PLANE  (HB * WB)      // 65536
#define OUT_W  2048
#define ALS_LD 66             // padded leading dim to avoid LDS bank conflicts
#define ST_LD  17             // padded staging leading dim

__device__ __forceinline__ float idct_basis(int x, int u) {
    const float PI = 3.14159265358979323846f;
    float cu = (u == 0) ? 0.70710678118654752f : 1.0f;
    return 0.5f * cu * __cosf((2.0f * x + 1.0f) * (float)u * PI / 16.0f);
}

__global__ __launch_bounds__(256) void idct_wmma_kernel(
        const float* __restrict__ dct,
        const float* __restrict__ meanv,
        const float* __restrict__ stdv,
        float* __restrict__ out) {
    // LDS: scaled basis matrix, bias vector, per-wave output staging
    __shared__ float Als[64][ALS_LD];        // (A (x) A)[m][k] * std[k] / 255
    __shared__ float biasP[64];              // ((A (x) A) @ mean + 128) / 255
    __shared__ float stage[8][64 * ST_LD];   // per-wave 64x16 result tile

    const int tid  = threadIdx.x;
    const int lane = tid & 31;
    const int wid  = tid >> 5;

    // ---- one-time LDS build (folds std, mean, +128, /255 into constants) ----
    for (int e = tid; e < 64 * 64; e += 256) {
        int m = e >> 6, k = e & 63;
        int x = m >> 3, y = m & 7, u = k >> 3, v = k & 7;
        Als[m][k] = idct_basis(x, u) * idct_basis(y, v) * stdv[k] * (1.0f / 255.0f);
    }
    if (tid < 64) {
        int m = tid, x = m >> 3, y = m & 7;
        float s = 128.0f;
        for (int k = 0; k < 64; ++k) {
            int u = k >> 3, v = k & 7;
            s += idct_basis(x, u) * idct_basis(y, v) * meanv[k];
        }
        biasP[m] = s * (1.0f / 255.0f);
    }
    __syncthreads();

    // ---- tile assignment: one 64x16 GEMM tile (16 consecutive w-blocks) per wave
    const int tile = blockIdx.x * 8 + wid;          // 0 .. 65535
    const int b    = tile >> 12;                    // 4096 tiles per batch
    const int rem  = tile & 4095;
    const int h    = rem >> 4;                      // block row 0..255
    const int w0   = (rem & 15) << 4;               // first of 16 w-blocks

    const int n    = lane & 15;                     // N index (w-block) 0..15
    const int half = lane >> 4;                     // selects K pair / M half

    // ---- load B fragments: D[k=c, n] ; B vgpr j holds K = ks*4 + half*2 + j
    const float* p = dct + (size_t)b * CH * PLANE + h * WB + w0 + n;
    float bfrag[16][2];
#pragma unroll
    for (int ks = 0; ks < 16; ++ks) {
        int c0 = ks * 4 + half * 2;
        bfrag[ks][0] = p[(size_t)c0 * PLANE];       // coalesced 64B per half-wave
        bfrag[ks][1] = p[(size_t)(c0 + 1) * PLANE];
    }

    float* st = stage[wid];
    const int mrow_lane = lane & 15;                // A fragment: lanes hold M = lane%16

#pragma unroll
    for (int mt = 0; mt < 4; ++mt) {
        // init accumulator from bias (includes +128 and /255 folding)
        v8f acc;
#pragma unroll
        for (int r = 0; r < 8; ++r)
            acc[r] = biasP[mt * 16 + half * 8 + r];

        const int mrow = mt * 16 + mrow_lane;
#pragma unroll
        for (int ks = 0; ks < 16; ++ks) {
            int kb = ks * 4 + half * 2;             // A vgpr j holds K = kb + j
            v2f afrag = *(const v2f*)&Als[mrow][kb]; // ds_load_b64
            v2f bb;
            bb.x = bfrag[ks][0];
            bb.y = bfrag[ks][1];
            acc = __builtin_amdgcn_wmma_f32_16x16x4_f32(
                      false, afrag, false, bb, (short)0, acc, false, false);
        }

        // stash C fragment in wave-private staging (layout: [m][n])
#pragma unroll
        for (int r = 0; r < 8; ++r)
            st[(mt * 16 + half * 8 + r) * ST_LD + n] = acc[r];
    }

    __syncthreads();

    // ---- coalesced epilogue: 8 image rows, 512B contiguous each -------------
    // output float index within row-segment: f = n*8 + y ; lane covers f = 4*lane..
    const int nn = lane >> 1;
    const int y0 = (lane & 1) * 4;
    const size_t obase = (size_t)b * OUT_W * OUT_W + (size_t)w0 * 8;
#pragma unroll
    for (int x = 0; x < 8; ++x) {
        float4 v;
        v.x = st[(x * 8 + y0 + 0) * ST_LD + nn];
        v.y = st[(x * 8 + y0 + 1) * ST_LD + nn];
        v.z = st[(x * 8 + y0 + 2) * ST_LD + nn];
        v.w = st[(x * 8 + y0 + 3) * ST_LD + nn];
        size_t off = obase + (size_t)(h * 8 + x) * OUT_W + (size_t)lane * 4;
        *reinterpret_cast<float4*>(out + off) = v;  // global_store_b128, 16B aligned
    }
}

extern "C" void kernel_launch(void* const* d_in, const int* in_sizes, int n_in,
                              void* d_out, int out_size, void* d_ws, size_t ws_size,
                              hipStream_t stream) {
    const float* dct   = (const float*)d_in[0];
    const float* meanv = (const float*)d_in[1];
    const float* stdv  = (const float*)d_in[2];
    float* out = (float*)d_out;

    // 16*256*256 pixel-blocks / 16 per tile / 8 waves per block = 8192 blocks
    dim3 grid(8192), block(256);
    idct_wmma_kernel<<<grid, block, 0, stream>>>(dct, meanv, stdv, out);
}